// PseudoLevinsonDurbinRecursion_17721035063364
// MI455X (gfx1250) — compile-verified
//
#include <hip/hip_runtime.h>
#include <math.h>

// ---------------------------------------------------------------------------
// PseudoLevinsonDurbinRecursion for MI455X (gfx1250, wave32)
//   B = 65536 rows, each row: r[0..24] -> out[0..24] = [g, a[0..23]]
//   Memory-bound: 13.1 MB total traffic -> ~0.56 us floor at 23.3 TB/s.
//   Plan: async global->LDS staging (ASYNCcnt path), per-lane fully unrolled
//   f32 Levinson recursion, g computed with chained V_WMMA_F32_16X16X4_F32
//   (batched r1.a dot products = diagonal of a 16x24 * 24x16 product).
// ---------------------------------------------------------------------------

#define M_ORDER 24
#define NCOEF   25
#define NTHREADS 256
#define ROWS_PER_BLOCK 256
#define FLOATS_PER_BLOCK (ROWS_PER_BLOCK * NCOEF)   // 6400
#define VEC4_PER_BLOCK   (FLOATS_PER_BLOCK / 4)     // 1600

typedef __attribute__((ext_vector_type(2))) float v2f;
typedef __attribute__((ext_vector_type(8))) float v8f;
typedef __attribute__((ext_vector_type(4))) int   v4i;

#define AS_GLOBAL __attribute__((address_space(1)))
#define AS_LDS    __attribute__((address_space(3)))

#if defined(__has_builtin)
#  if __has_builtin(__builtin_amdgcn_wmma_f32_16x16x4_f32)
#    define HAVE_WMMA_F32X4 1
#  endif
#  if __has_builtin(__builtin_amdgcn_global_load_async_to_lds_b128) && \
      __has_builtin(__builtin_amdgcn_s_wait_asynccnt)
#    define HAVE_ASYNC_LD 1
#  endif
#  if __has_builtin(__builtin_amdgcn_global_store_async_from_lds_b128) && \
      __has_builtin(__builtin_amdgcn_s_wait_asynccnt)
#    define HAVE_ASYNC_ST 1
#  endif
#endif

__global__ __launch_bounds__(NTHREADS)
void levinson_durbin_kernel(const float* __restrict__ g_r,
                            float* __restrict__ g_out)
{
    __shared__ __align__(16) float lds_in [FLOATS_PER_BLOCK];
    __shared__ __align__(16) float lds_out[FLOATS_PER_BLOCK];

    const int tid = threadIdx.x;
    const float* gin  = g_r  + (size_t)blockIdx.x * FLOATS_PER_BLOCK;
    float*       gout = g_out + (size_t)blockIdx.x * FLOATS_PER_BLOCK;

    // ---- Stage 1: coalesced global -> LDS (b128 async-to-LDS path) --------
#if HAVE_ASYNC_LD
    #pragma unroll
    for (int k = 0; k < 6; ++k) {
        const int i = tid + k * NTHREADS;
        __builtin_amdgcn_global_load_async_to_lds_b128(
            (AS_GLOBAL v4i*)(gin + 4 * i),
            (AS_LDS    v4i*)(lds_in + 4 * i), 0, 0);
    }
    if (tid < VEC4_PER_BLOCK - 6 * NTHREADS) {
        const int i = tid + 6 * NTHREADS;
        __builtin_amdgcn_global_load_async_to_lds_b128(
            (AS_GLOBAL v4i*)(gin + 4 * i),
            (AS_LDS    v4i*)(lds_in + 4 * i), 0, 0);
    }
    __builtin_amdgcn_s_wait_asynccnt(0);
#else
    {
        const float4* gin4 = reinterpret_cast<const float4*>(gin);
        float4* lin4 = reinterpret_cast<float4*>(lds_in);
        #pragma unroll
        for (int k = 0; k < 6; ++k) {
            const int i = tid + k * NTHREADS;
            lin4[i] = gin4[i];
        }
        if (tid < VEC4_PER_BLOCK - 6 * NTHREADS) {
            const int i = tid + 6 * NTHREADS;
            lin4[i] = gin4[i];
        }
    }
#endif
    __syncthreads();

    // ---- Stage 2: per-lane Levinson-Durbin (fully unrolled, registers) ----
    // Row stride 25, gcd(25,64)=1 -> conflict-free LDS reads across a wave.
    const float* rrow = lds_in + tid * NCOEF;
    float rr[NCOEF];
    #pragma unroll
    for (int i = 0; i < NCOEF; ++i) rr[i] = rrow[i];

    float a[NCOEF];           // a[1..24] valid
    float E = rr[0];
    #pragma unroll
    for (int m = 1; m <= M_ORDER; ++m) {
        float acc = rr[m];
        #pragma unroll
        for (int i = 1; i < m; ++i) acc = fmaf(a[i], rr[m - i], acc);
        const float kref = -acc / E;
        #pragma unroll
        for (int i = 1; 2 * i < m; ++i) {
            const float ai = a[i], aj = a[m - i];
            a[i]     = fmaf(kref, aj, ai);
            a[m - i] = fmaf(kref, ai, aj);
        }
        if ((m & 1) == 0) a[m >> 1] = a[m >> 1] * (1.0f + kref);
        a[m] = kref;
        E = fmaf(-kref * kref, E, E);
    }

    float* orow = lds_out + tid * NCOEF;
    #pragma unroll
    for (int i = 1; i <= M_ORDER; ++i) orow[i] = a[i];
    __syncthreads();

    // ---- Stage 3: g = sqrt(r1.a + r0) via chained WMMA f32 16x16x4 --------
    // For each 16-row group of this wave: D = A(16x24 of r1) * B(24x16 of a);
    // diagonal D[j][j] = r1_j . a_j. K=24 -> 6 chained WMMAs.
    const int lane     = tid & 31;
    const int waveRow0 = tid & ~31;
#if HAVE_WMMA_F32X4
    const int khi = (lane >> 4) * 2;                      // 0 or 2
    #pragma unroll
    for (int grp = 0; grp < 2; ++grp) {
        const int rb = waveRow0 + grp * 16;
        const int fr = rb + (lane & 15);     // fragment row for this lane
        v8f c = {};
        #pragma unroll
        for (int kb = 0; kb < 6; ++kb) {
            const int k0 = kb * 4 + khi;     // 0-based k within r1 / a
            v2f A, Bv;
            // A: 16x4 slice of r1 rows (r1[j,k] = r[j,k+1])
            A.x  = lds_in [fr * NCOEF + 1 + k0 + 0];
            A.y  = lds_in [fr * NCOEF + 1 + k0 + 1];
            // B: 4x16 slice, B[k][n] = a_rown[k]
            Bv.x = lds_out[fr * NCOEF + 1 + k0 + 0];
            Bv.y = lds_out[fr * NCOEF + 1 + k0 + 1];
            c = __builtin_amdgcn_wmma_f32_16x16x4_f32(
                    false, A, false, Bv, (short)0, c, false, false);
        }
        // Diagonal extraction: D[v][v] at lane v (VGPR v), D[v+8][v+8] at
        // lane v+24 (VGPR v), per the 16x16 f32 C/D layout.
        #pragma unroll
        for (int v = 0; v < 8; ++v) {
            int j = -1;
            if (lane == v)      j = v;
            if (lane == v + 24) j = v + 8;
            if (j >= 0) {
                const int row2 = rb + j;
                const float g2 = c[v] + lds_in[row2 * NCOEF];
                lds_out[row2 * NCOEF] = sqrtf(g2);
            }
        }
    }
#else
    // Scalar fallback: per-lane dot product.
    {
        float acc = rr[0];
        #pragma unroll
        for (int i = 1; i <= M_ORDER; ++i) acc = fmaf(rr[i], a[i], acc);
        orow[0] = sqrtf(acc);
    }
#endif
    __syncthreads();

    // ---- Stage 4: coalesced LDS -> global (b128 async-from-LDS path) ------
#if HAVE_ASYNC_ST
    #pragma unroll
    for (int k = 0; k < 6; ++k) {
        const int i = tid + k * NTHREADS;
        __builtin_amdgcn_global_store_async_from_lds_b128(
            (AS_GLOBAL v4i*)(gout + 4 * i),
            (AS_LDS    v4i*)(lds_out + 4 * i), 0, 0);
    }
    if (tid < VEC4_PER_BLOCK - 6 * NTHREADS) {
        const int i = tid + 6 * NTHREADS;
        __builtin_amdgcn_global_store_async_from_lds_b128(
            (AS_GLOBAL v4i*)(gout + 4 * i),
            (AS_LDS    v4i*)(lds_out + 4 * i), 0, 0);
    }
    __builtin_amdgcn_s_wait_asynccnt(0);
#else
    {
        float4* gout4 = reinterpret_cast<float4*>(gout);
        const float4* lout4 = reinterpret_cast<const float4*>(lds_out);
        #pragma unroll
        for (int k = 0; k < 6; ++k) {
            const int i = tid + k * NTHREADS;
            gout4[i] = lout4[i];
        }
        if (tid < VEC4_PER_BLOCK - 6 * NTHREADS) {
            const int i = tid + 6 * NTHREADS;
            gout4[i] = lout4[i];
        }
    }
#endif
}

extern "C" void kernel_launch(void* const* d_in, const int* in_sizes, int n_in,
                              void* d_out, int out_size, void* d_ws, size_t ws_size,
                              hipStream_t stream) {
    const float* r = (const float*)d_in[0];   // (B, 25) float32
    // d_in[1] is n_out == 1 (scalar); output is concat(g, a) -> (B, 25)
    float* out = (float*)d_out;
    const int nrows  = in_sizes[0] / NCOEF;          // 65536
    const int blocks = nrows / ROWS_PER_BLOCK;       // 256
    levinson_durbin_kernel<<<blocks, NTHREADS, 0, stream>>>(r, out);
}